// TTAWarper_11982958756190
// MI455X (gfx1250) — compile-verified
//
#include <hip/hip_runtime.h>
#include <stdint.h>

#define NBOX     2048
#define BUFROWS  4096      // max emitted rows = 2*N (proven bound)
#define MAXDET   100
#define NTHREADS 1024      // 32 wave32s, one WGP, fully LDS-resident

typedef float v2f  __attribute__((ext_vector_type(2)));
typedef float v8f  __attribute__((ext_vector_type(8)));
typedef unsigned int u32x4 __attribute__((ext_vector_type(4)));
typedef int   i32x8 __attribute__((ext_vector_type(8)));
typedef int   i32x4 __attribute__((ext_vector_type(4)));

// Sum the 64 values {p0,p1} held across one wave (2 per lane) using
// V_WMMA_F32_16X16X4_F32 with A == ones(16x4): every D entry is a column sum
// of B, so summing the 16 columns (4 shuffle steps on D vgpr0) yields the
// total of all 64 B slots regardless of the exact B lane layout. Exact f32 FMA.
__device__ __forceinline__ float wave_sum64(float p0, float p1) {
#if __has_builtin(__builtin_amdgcn_wmma_f32_16x16x4_f32)
  v2f a; a.x = 1.0f; a.y = 1.0f;
  v2f b; b.x = p0;   b.y = p1;
  v8f c; c[0]=0.f;c[1]=0.f;c[2]=0.f;c[3]=0.f;c[4]=0.f;c[5]=0.f;c[6]=0.f;c[7]=0.f;
  c = __builtin_amdgcn_wmma_f32_16x16x4_f32(false, a, false, b, (short)0, c, false, false);
  float v = c[0];                       // D[m][n]: lanes hold colsum[lane&15]
  v += __shfl_xor(v, 1, 32);
  v += __shfl_xor(v, 2, 32);
  v += __shfl_xor(v, 4, 32);
  v += __shfl_xor(v, 8, 32);            // each 16-lane half now holds full total
  return v;
#else
  float v = p0 + p1;
  v += __shfl_xor(v, 1, 32);
  v += __shfl_xor(v, 2, 32);
  v += __shfl_xor(v, 4, 32);
  v += __shfl_xor(v, 8, 32);
  v += __shfl_xor(v, 16, 32);
  return v;
#endif
}

// In-LDS bitonic sort: descending score, ties broken by ascending index
// (== stable jnp.argsort(-score)). All records distinct via index -> total order.
__device__ void bitonic_desc(float* key, int* idx, int n, int tid) {
  for (int k = 2; k <= n; k <<= 1) {
    for (int j = k >> 1; j > 0; j >>= 1) {
      __syncthreads();
      for (int i = tid; i < n; i += NTHREADS) {
        int ixj = i ^ j;
        if (ixj > i) {
          float ka = key[i], kb = key[ixj];
          int   ia = idx[i], ib = idx[ixj];
          bool beforeAB = (ka > kb) || (ka == kb && ia < ib);
          bool asc = ((i & k) == 0);
          bool doSwap = asc ? (!beforeAB) : beforeAB;
          if (doSwap) { key[i]=kb; key[ixj]=ka; idx[i]=ib; idx[ixj]=ia; }
        }
      }
    }
  }
  __syncthreads();
}

extern "C" __global__ __launch_bounds__(NTHREADS, 1)
void bbox_vote_soft_kernel(const float* __restrict__ gBoxes,
                           const float* __restrict__ gScores,
                           const int*   __restrict__ gLabels,
                           float*       __restrict__ out)
{
  const int tid  = threadIdx.x;
  const int lane = tid & 31;
  const int wv   = tid >> 5;

  // ---- LDS layout (~201 KB of the 320 KB WGP LDS) ----
  __shared__ float bufmem[6 * BUFROWS];   // SoA output rows; first 8192 f also = TDM stage for boxes
  __shared__ float detm[6 * NBOX];        // sorted det: x1,y1,x2,y2,s,lab
  __shared__ float areaLds[NBOX];
  __shared__ float skey[BUFROWS];
  __shared__ int   sidx[BUFROWS];
  __shared__ int   ilab[NBOX];
  __shared__ int   aliveLds[NBOX];
  __shared__ float wsums[5 * 32];
  __shared__ int   softCnt[32];
  __shared__ int   waveOff[32];
  __shared__ float redbuf[32];
  __shared__ float maxcLds;
  __shared__ int   headLds, cntLds;

  // ---- Stage boxes (8192 f32 = 32KB) into LDS via the Tensor Data Mover ----
#if __has_builtin(__builtin_amdgcn_tensor_load_to_lds) && __has_builtin(__builtin_amdgcn_s_wait_tensorcnt)
  if (tid < 32) {   // wave 0 issues one DMA (EXEC ignored by TDM)
    uint64_t ga  = (uint64_t)(uintptr_t)gBoxes;
    unsigned lds = (unsigned)(uintptr_t)(void*)&bufmem[0];  // low 32b of generic = LDS byte offset
    u32x4 g0;
    g0.x = 1u;                                   // count=1, user descriptor
    g0.y = lds;                                  // lds_addr (bytes)
    g0.z = (unsigned)(ga & 0xffffffffu);         // global_addr[31:0]
    g0.w = (unsigned)((ga >> 32) & 0x1ffffffu) | (2u << 30);  // addr[56:32] | type=2
    i32x8 g1;
    g1[0] = (int)(2u << 16);                     // data_size=4B; no multicast/pad/iterate
    g1[1] = (int)(((unsigned)(NBOX * 4)) << 16); // tensor_dim0 = 8192 (low 16 @ bits 63:48)
    g1[2] = (int)(1u << 16);                     // tensor_dim1 = 1
    g1[3] = (int)(((unsigned)(NBOX * 4)) << 16); // tile_dim0 = 8192
    g1[4] = 1;                                   // tile_dim1 = 1, tile_dim2 = 0
    g1[5] = NBOX * 4;                            // tensor_dim0_stride = 8192
    g1[6] = 0;
    g1[7] = 0;
    i32x4 z4; z4[0]=0; z4[1]=0; z4[2]=0; z4[3]=0;
    i32x8 z8; z8[0]=0; z8[1]=0; z8[2]=0; z8[3]=0; z8[4]=0; z8[5]=0; z8[6]=0; z8[7]=0;
    // 6-arg variant (clang-23 / therock-10.0 headers): (g0, g1, g2, g3, g4, cpol)
    __builtin_amdgcn_tensor_load_to_lds(g0, g1, z4, z4, z8, 0);
  }
#else
  for (int i = tid; i < NBOX * 4; i += NTHREADS) bufmem[i] = gBoxes[i];
#endif

  // Meanwhile: scores + labels into LDS, sort keys initialized
  for (int i = tid; i < NBOX; i += NTHREADS) {
    skey[i] = gScores[i];
    sidx[i] = i;
    ilab[i] = gLabels[i];
  }
#if __has_builtin(__builtin_amdgcn_tensor_load_to_lds) && __has_builtin(__builtin_amdgcn_s_wait_tensorcnt)
  if (tid < 32) __builtin_amdgcn_s_wait_tensorcnt(0);
#endif
  __syncthreads();

  // ---- maxc = max(boxes) + 1 ----
  float mx = -1e30f;
  for (int i = tid; i < NBOX * 4; i += NTHREADS) mx = fmaxf(mx, bufmem[i]);
  for (int off = 16; off > 0; off >>= 1) mx = fmaxf(mx, __shfl_xor(mx, off, 32));
  if (lane == 0) redbuf[wv] = mx;
  __syncthreads();
  if (tid == 0) {
    float m = redbuf[0];
    for (int w = 1; w < 32; ++w) m = fmaxf(m, redbuf[w]);
    maxcLds = m + 1.0f;
  }
  __syncthreads();
  const float maxc = maxcLds;

  // ---- sort detections by score desc (stable) ----
  bitonic_desc(skey, sidx, NBOX, tid);

  // ---- build sorted det (class-offset coords) + area, init alive ----
  for (int i = tid; i < NBOX; i += NTHREADS) {
    int o = sidx[i];
    float lab = (float)ilab[o];
    float off = lab * maxc;
    float x1 = bufmem[4*o+0] + off;
    float y1 = bufmem[4*o+1] + off;
    float x2 = bufmem[4*o+2] + off;
    float y2 = bufmem[4*o+3] + off;
    detm[0*NBOX+i] = x1; detm[1*NBOX+i] = y1;
    detm[2*NBOX+i] = x2; detm[3*NBOX+i] = y2;
    detm[4*NBOX+i] = skey[i];
    detm[5*NBOX+i] = lab;
    areaLds[i] = (x2 - x1) * (y2 - y1);
    aliveLds[i] = 1;
  }
  if (tid == 0) { headLds = 0; cntLds = 0; }
  __syncthreads();

  // ---- greedy vote loop: 2 barriers per iteration, head strictly increases ----
  const int e0 = 2 * tid, e1 = 2 * tid + 1;
  for (int it = 0; it <= NBOX; ++it) {
    int head = headLds;
    if (head < 0) break;

    // Phase A: everyone computes IoU vs cur, partials, soft flags
    float cx1 = detm[0*NBOX+head], cy1 = detm[1*NBOX+head];
    float cx2 = detm[2*NBOX+head], cy2 = detm[3*NBOX+head];
    float cs  = detm[4*NBOX+head], clab = detm[5*NBOX+head];
    float carea = areaLds[head];

    float x10 = detm[e0], y10 = detm[NBOX+e0], x20 = detm[2*NBOX+e0], y20 = detm[3*NBOX+e0];
    float s0v = detm[4*NBOX+e0];
    float i0x = fmaxf(fminf(cx2, x20) - fmaxf(cx1, x10), 0.f);
    float i0y = fmaxf(fminf(cy2, y20) - fmaxf(cy1, y10), 0.f);
    float inter0 = i0x * i0y;
    float iou0 = inter0 / (carea + areaLds[e0] - inter0);
    bool mg0 = (aliveLds[e0] != 0) && (iou0 >= 0.65f);
    float m0 = mg0 ? s0v : 0.f;
    float ss0 = s0v * (1.f - iou0);
    bool sv0 = mg0 && (ss0 >= 0.05f);   // n_merge>1 gate provably redundant

    float x11 = detm[e1], y11 = detm[NBOX+e1], x21 = detm[2*NBOX+e1], y21 = detm[3*NBOX+e1];
    float s1v = detm[4*NBOX+e1];
    float i1x = fmaxf(fminf(cx2, x21) - fmaxf(cx1, x11), 0.f);
    float i1y = fmaxf(fminf(cy2, y21) - fmaxf(cy1, y11), 0.f);
    float inter1 = i1x * i1y;
    float iou1 = inter1 / (carea + areaLds[e1] - inter1);
    bool mg1 = (aliveLds[e1] != 0) && (iou1 >= 0.65f);
    float m1 = mg1 ? s1v : 0.f;
    float ss1 = s1v * (1.f - iou1);
    bool sv1 = mg1 && (ss1 >= 0.05f);

    // WMMA-based per-wave weighted sums (5 quantities)
    float t;
    t = wave_sum64(m0 * x10, m1 * x11); if (lane == 0) wsums[0*32+wv] = t;
    t = wave_sum64(m0 * y10, m1 * y11); if (lane == 0) wsums[1*32+wv] = t;
    t = wave_sum64(m0 * x20, m1 * x21); if (lane == 0) wsums[2*32+wv] = t;
    t = wave_sum64(m0 * y20, m1 * y21); if (lane == 0) wsums[3*32+wv] = t;
    t = wave_sum64(m0,       m1      ); if (lane == 0) wsums[4*32+wv] = t;

    // Deterministic soft-row compaction: wave32 ballot prefix (index order)
    unsigned b0 = (unsigned)__ballot((int)sv0);
    unsigned b1 = (unsigned)__ballot((int)sv1);
    unsigned mask = (1u << lane) - 1u;
    int pre = __popc(b0 & mask) + __popc(b1 & mask);
    int idx0 = pre, idx1 = pre + (sv0 ? 1 : 0);
    if (lane == 0) softCnt[wv] = __popc(b0) + __popc(b1);
    if (mg0) aliveLds[e0] = 0;
    if (mg1) aliveLds[e1] = 0;
    __syncthreads();

    // Phase B: wave 0 reduces across waves + writes row0; wave 1 advances head
    if (tid < 32) {
      int oldCnt = cntLds;
      float q0 = wsums[tid], q1 = wsums[32+tid], q2 = wsums[64+tid],
            q3 = wsums[96+tid], q4 = wsums[128+tid];
      for (int off = 16; off > 0; off >>= 1) {
        q0 += __shfl_xor(q0, off, 32);
        q1 += __shfl_xor(q1, off, 32);
        q2 += __shfl_xor(q2, off, 32);
        q3 += __shfl_xor(q3, off, 32);
        q4 += __shfl_xor(q4, off, 32);
      }
      if (tid == 0) {
        float wsafe = (q4 > 0.f) ? q4 : 1.f;
        bufmem[0*BUFROWS+oldCnt] = q0 / wsafe;
        bufmem[1*BUFROWS+oldCnt] = q1 / wsafe;
        bufmem[2*BUFROWS+oldCnt] = q2 / wsafe;
        bufmem[3*BUFROWS+oldCnt] = q3 / wsafe;
        bufmem[4*BUFROWS+oldCnt] = cs;     // maxs == s[head] (sorted, head in merge)
        bufmem[5*BUFROWS+oldCnt] = clab;
      }
      int c = softCnt[tid];
      int inc = c;
      for (int off = 1; off < 32; off <<= 1) {
        int u = __shfl_up(inc, off, 32);
        if (tid >= off) inc += u;
      }
      waveOff[tid] = oldCnt + 1 + (inc - c);     // exclusive scan + row0 slot
      if (tid == 31) cntLds = oldCnt + 1 + inc;  // lane31 holds grand total
    } else if (tid == 32) {
      int h = head;
      while (h < NBOX && aliveLds[h] == 0) ++h;  // O(N) total over whole loop
      headLds = (h < NBOX) ? h : -1;
    }
    __syncthreads();

    // Phase C: emit soft rows at deterministic positions
    if (sv0) {
      int r = waveOff[wv] + idx0;
      bufmem[r] = x10; bufmem[BUFROWS+r] = y10;
      bufmem[2*BUFROWS+r] = x20; bufmem[3*BUFROWS+r] = y20;
      bufmem[4*BUFROWS+r] = ss0; bufmem[5*BUFROWS+r] = detm[5*NBOX+e0];
    }
    if (sv1) {
      int r = waveOff[wv] + idx1;
      bufmem[r] = x11; bufmem[BUFROWS+r] = y11;
      bufmem[2*BUFROWS+r] = x21; bufmem[3*BUFROWS+r] = y21;
      bufmem[4*BUFROWS+r] = ss1; bufmem[5*BUFROWS+r] = detm[5*NBOX+e1];
    }
    // no barrier needed: next Phase A touches disjoint LDS; sync1/sync2 order all hazards
  }
  __syncthreads();

  // ---- final top-100 by score (stable) ----
  int cnt = cntLds;
  for (int i = tid; i < BUFROWS; i += NTHREADS) {
    skey[i] = (i < cnt) ? bufmem[4*BUFROWS+i] : -1e30f;
    sidx[i] = i;
  }
  __syncthreads();
  bitonic_desc(skey, sidx, BUFROWS, tid);

  if (tid < MAXDET) {
    int r = tid;
    float sc = skey[r];
    bool valid = sc > -1e29f;   // real scores >= SOFT_TH
    int bi = sidx[r];
    float lab = valid ? bufmem[5*BUFROWS+bi] : 0.f;
    float off = lab * maxc;
    out[4*r + 0] = valid ? bufmem[0*BUFROWS+bi] - off : 0.f;
    out[4*r + 1] = valid ? bufmem[1*BUFROWS+bi] - off : 0.f;
    out[4*r + 2] = valid ? bufmem[2*BUFROWS+bi] - off : 0.f;
    out[4*r + 3] = valid ? bufmem[3*BUFROWS+bi] - off : 0.f;
    out[4*MAXDET + r] = valid ? sc : 0.f;
    out[5*MAXDET + r] = lab;
  }
}

extern "C" void kernel_launch(void* const* d_in, const int* in_sizes, int n_in,
                              void* d_out, int out_size, void* d_ws, size_t ws_size,
                              hipStream_t stream) {
  (void)in_sizes; (void)n_in; (void)out_size; (void)d_ws; (void)ws_size;
  const float* boxes  = (const float*)d_in[0];
  const float* scores = (const float*)d_in[1];
  const int*   labels = (const int*)d_in[2];
  float* out = (float*)d_out;
  bbox_vote_soft_kernel<<<1, NTHREADS, 0, stream>>>(boxes, scores, labels, out);
}